// Interpolater_43344809952128
// MI455X (gfx1250) — compile-verified
//
#include <hip/hip_runtime.h>
#include <hip/hip_bf16.h>

typedef __attribute__((ext_vector_type(4))) float f4;

#ifndef USE_ASYNC_LDS
#define USE_ASYNC_LDS 1
#endif

namespace {
constexpr int kH = 1024;
constexpr int kW = 1024;
constexpr int kC = 32;
constexpr int kP = 65536;          // points per image (N*P = in_sizes[1])
constexpr int kPtsPerBlock = 32;   // 8 lanes per point, 256 threads
constexpr int kThreads = 256;
static_assert(kP == (1 << 16), "pt>>16 assumes P==65536");
}  // namespace

// Address-space-qualified pointer types for the async global->LDS builtin.
// (clang diagnostic showed param 0 is a non-const AS1 'int*'.)
typedef __attribute__((address_space(1))) int g1int;
typedef __attribute__((address_space(3))) int l3int;

// One block = 32 points. Lanes 0..7 of a point each own a float4 channel-quad
// (32 channels * 4B = 128B per tap -> one coalesced 128B request per tap).
__global__ __launch_bounds__(kThreads) void bilerp_gather_kernel(
    const float* __restrict__ data,   // (N,H,W,C) f32
    const float* __restrict__ sub_x,  // (N,P) f32
    const float* __restrict__ sub_y,  // (N,P) f32
    float* __restrict__ out)          // (N,P,C) f32
{
  __shared__ float s_xy[2 * kPtsPerBlock];  // [0,32): sx  [32,64): sy

  const int tid = threadIdx.x;
  const int pt0 = blockIdx.x * kPtsPerBlock;

  // ---- Stage coordinates into LDS via CDNA5 async global->LDS copy ----
  // Wave 0 pulls sub_x, wave 1 pulls sub_y (full-EXEC waves).
#if USE_ASYNC_LDS && __has_builtin(__builtin_amdgcn_global_load_async_to_lds_b32)
  if (tid < kPtsPerBlock) {
    __builtin_amdgcn_global_load_async_to_lds_b32(
        (g1int*)(sub_x + pt0 + tid),
        (l3int*)&s_xy[tid],
        /*offset=*/0, /*cpol=*/0);
  } else if (tid < 2 * kPtsPerBlock) {
    __builtin_amdgcn_global_load_async_to_lds_b32(
        (g1int*)(sub_y + pt0 + (tid - kPtsPerBlock)),
        (l3int*)&s_xy[tid],
        /*offset=*/0, /*cpol=*/0);
  }
#if __has_builtin(__builtin_amdgcn_s_wait_asynccnt)
  __builtin_amdgcn_s_wait_asynccnt(0);
#else
  asm volatile("s_wait_asynccnt 0" ::: "memory");
#endif
#else
  if (tid < kPtsPerBlock) {
    s_xy[tid] = sub_x[pt0 + tid];
  } else if (tid < 2 * kPtsPerBlock) {
    s_xy[tid] = sub_y[pt0 + (tid - kPtsPerBlock)];
  }
#endif
  __syncthreads();

  const int lp = tid >> 3;          // local point 0..31
  const int cq = (tid & 7) << 2;    // channel quad: 0,4,...,28
  const int pt = pt0 + lp;          // global point index = n*P + p
  const int n  = pt >> 16;          // pt / kP

  const float sx = s_xy[lp];
  const float sy = s_xy[kPtsPerBlock + lp];

  const float x0f = floorf(sx);
  const float y0f = floorf(sy);
  const float wx = sx - x0f;
  const float wy = sy - y0f;

  int x0 = min(max((int)x0f, 0), kW - 1);
  int x1 = min(max((int)x0f + 1, 0), kW - 1);
  int y0 = min(max((int)y0f, 0), kH - 1);
  int y1 = min(max((int)y0f + 1, 0), kH - 1);

  const size_t img = (size_t)n * ((size_t)kH * kW * kC);
  const float* b00 = data + img + (((size_t)y0 * kW + x0) * kC) + cq;
  const float* b01 = data + img + (((size_t)y0 * kW + x1) * kC) + cq;
  const float* b10 = data + img + (((size_t)y1 * kW + x0) * kC) + cq;
  const float* b11 = data + img + (((size_t)y1 * kW + x1) * kC) + cq;

  // Single-use gather traffic: non-temporal so we don't thrash the 192MB L2.
  f4 v00 = __builtin_nontemporal_load((const f4*)b00);
  f4 v01 = __builtin_nontemporal_load((const f4*)b01);
  f4 v10 = __builtin_nontemporal_load((const f4*)b10);
  f4 v11 = __builtin_nontemporal_load((const f4*)b11);

  const float w00 = (1.0f - wx) * (1.0f - wy);
  const float w01 = wx * (1.0f - wy);
  const float w10 = (1.0f - wx) * wy;
  const float w11 = wx * wy;

  f4 r = v00 * w00 + v01 * w01 + v10 * w10 + v11 * w11;

  // Streaming output: non-temporal store.
  __builtin_nontemporal_store(r, (f4*)(out + (size_t)pt * kC + cq));
}

extern "C" void kernel_launch(void* const* d_in, const int* in_sizes, int n_in,
                              void* d_out, int out_size, void* d_ws, size_t ws_size,
                              hipStream_t stream) {
  const float* data  = (const float*)d_in[0];
  const float* sub_x = (const float*)d_in[1];
  const float* sub_y = (const float*)d_in[2];
  float* out = (float*)d_out;

  const int total_pts = in_sizes[1];  // N*P = 524288
  const int blocks = (total_pts + kPtsPerBlock - 1) / kPtsPerBlock;  // 16384

  bilerp_gather_kernel<<<dim3(blocks), dim3(kThreads), 0, stream>>>(
      data, sub_x, sub_y, out);
}